// FAENet_82841329205727
// MI455X (gfx1250) — compile-verified
//
#include <hip/hip_runtime.h>
#include <math.h>

#define N_NODES 50000
#define N_EDGES 800000
#define N_GRAPH 64
#define HDIM    128
#define LLAYERS 4
#define KIN     384            // NF + 2*H
#define EPSN    1e-5f

#define APAD  388              // A-tile LDS row stride (f32): 388%64=4 -> conflict-free b64 frag loads
#define APAD2 132              // node-gemm A stride: 132%64=4
#define BPAD  136              // Bt LDS row stride: 136%64=8 -> conflict-free b32 frag loads

typedef float v2f __attribute__((ext_vector_type(2)));
typedef float v8f __attribute__((ext_vector_type(8)));

// fast swish: x * sigmoid(x) = x * rcp(1 + exp(-x)); v_exp_f32 + v_rcp_f32 (no IEEE div expansion)
__device__ __forceinline__ float swishf(float x) {
  return x * __builtin_amdgcn_rcpf(1.0f + __expf(-x));
}

// wave-synchronous LDS fence: hardware wait + compiler memory barrier
__device__ __forceinline__ void lds_fence() { asm volatile("s_wait_dscnt 0" ::: "memory"); }

__device__ __forceinline__ float wave_sum32(float v) {
#pragma unroll
  for (int o = 16; o > 0; o >>= 1) v += __shfl_xor(v, o, 32);
  return v;
}

// ---------------------------------------------------------------- utilities
__global__ void zero_kernel(float* __restrict__ p, int n) {
  for (int i = blockIdx.x * blockDim.x + threadIdx.x; i < n; i += gridDim.x * blockDim.x)
    p[i] = 0.0f;
}

__global__ void copy_kernel(float* __restrict__ dst, const float* __restrict__ src, int n) {
  for (int i = blockIdx.x * blockDim.x + threadIdx.x; i < n; i += gridDim.x * blockDim.x)
    dst[i] = src[i];
}

// dst[l][k][n] = src[l][n][k]   (src row-major [layers][128][K])
__global__ void transpose_w_kernel(float* __restrict__ dst, const float* __restrict__ src,
                                   int layers, int K) {
  int total = layers * K * HDIM;
  for (int i = blockIdx.x * blockDim.x + threadIdx.x; i < total; i += gridDim.x * blockDim.x) {
    int l = i / (K * HDIM);
    int r = i - l * K * HDIM;
    int k = r >> 7;
    int n = r & 127;
    dst[i] = src[l * K * HDIM + n * K + k];
  }
}

// ---------------------------------------------------------------- embedding
__global__ void embed_h_kernel(const float* __restrict__ forces,
                               const float* __restrict__ w_h1, const float* __restrict__ b_h1,
                               const float* __restrict__ w_h12, const float* __restrict__ b_h12,
                               float* __restrict__ h) {
  int total = N_NODES * HDIM;
  for (int i = blockIdx.x * blockDim.x + threadIdx.x; i < total; i += gridDim.x * blockDim.x) {
    int n = i >> 7, c = i & 127;
    float fx = forces[n * 3], fy = forces[n * 3 + 1], fz = forces[n * 3 + 2];
    float v;
    if (c < 64) {
      v = fx * w_h1[c * 3] + fy * w_h1[c * 3 + 1] + fz * w_h1[c * 3 + 2] + b_h1[c];
    } else {
      int cc = c - 64;
      float fn = sqrtf(fx * fx + fy * fy + fz * fz);
      v = w_h12[cc] * fn + b_h12[cc];
    }
    h[i] = swishf(v);
  }
}

// ---------------------------------------------------------------- output block
__global__ __launch_bounds__(256) void out_block_kernel(
    const float* __restrict__ h, const int* __restrict__ batch,
    const float* __restrict__ w_wlin, const float* __restrict__ b_wlin,
    const float* __restrict__ w_lin1, const float* __restrict__ b_lin1,
    const float* __restrict__ w_lin2, const float* __restrict__ b_lin2,
    float* __restrict__ energy, int layer) {
  __shared__ float sh[8 * 128];
  const int tid = threadIdx.x, wave = tid >> 5, lane = tid & 31;
  const int stride = gridDim.x * 8;
  for (int n = blockIdx.x * 8 + wave; n < N_NODES; n += stride) {
#pragma unroll
    for (int k = 0; k < 4; ++k) sh[wave * 128 + lane + 32 * k] = h[n * 128 + lane + 32 * k];
    lds_fence();
    float ap = 0.0f;
#pragma unroll
    for (int k = 0; k < 4; ++k) ap += sh[wave * 128 + lane + 32 * k] * w_wlin[lane + 32 * k];
    ap = wave_sum32(ap) + b_wlin[0];
    float p2 = 0.0f;
#pragma unroll
    for (int j = 0; j < 2; ++j) {
      int cc = lane + 32 * j;
      float a = b_lin1[cc];
      for (int c = 0; c < 128; ++c) a += sh[wave * 128 + c] * w_lin1[cc * 128 + c];
      p2 += swishf(a) * w_lin2[cc];
    }
    float t2 = wave_sum32(p2) + b_lin2[0];
    if (lane == 0) atomicAdd(&energy[batch[n] * (LLAYERS + 1) + layer], t2 * ap);
  }
}

// ---------------------------------------------------------------- edge GEMM + scatter (WMMA)
// per wave: 16-edge tile.  A = [e | h[src] | h[dst]] (16x384, e recomputed), B = w_geom^T in LDS.
// D = swish(A@B + bias);  h_out[dst] += h[src] * D   (fused atomic scatter)
__global__ __launch_bounds__(64) void edge_gemm_scatter(
    const float* __restrict__ h_in, float* __restrict__ h_out,
    const float* __restrict__ pos, const float* __restrict__ beam,
    const int* __restrict__ src, const int* __restrict__ dst,
    const float* __restrict__ Bt, const float* __restrict__ bias,
    const float* __restrict__ w_e1, const float* __restrict__ b_e1,
    const float* __restrict__ w_e12, const float* __restrict__ b_e12) {
  extern __shared__ char smemraw[];
  float* sB   = (float*)smemraw;              // KIN*BPAD
  float* sA   = sB + KIN * BPAD;              // 2*16*APAD
  int*   sSrc = (int*)(sA + 2 * 16 * APAD);   // 32
  int*   sDst = sSrc + 32;                    // 32
  float* sAtt = (float*)(sDst + 32);          // 2*16*8

  const int tid = threadIdx.x;
  const int wave = tid >> 5;
  const int lane = tid & 31;

  for (int i = tid; i < KIN * HDIM; i += 64) {
    int k = i >> 7, n = i & 127;
    sB[k * BPAD + n] = Bt[i];
  }
  __syncthreads();   // one pre-loop block barrier (all threads reach exactly once)

  float* sAw  = sA + wave * (16 * APAD);
  int*   srcw = sSrc + wave * 16;
  int*   dstw = sDst + wave * 16;
  float* attw = sAtt + wave * 16 * 8;

  const int m16   = lane & 15;
  const int khalf = lane >> 4;
  const int ntiles = N_EDGES / 16;

  for (int tile = blockIdx.x * 2 + wave; tile < ntiles; tile += gridDim.x * 2) {
    if (lane < 16) {
      int eid = tile * 16 + lane;
      int s = src[eid], d = dst[eid];
      srcw[lane] = s;
      dstw[lane] = d;
      float rx = pos[3 * s] - pos[3 * d];
      float ry = pos[3 * s + 1] - pos[3 * d + 1];
      float rz = pos[3 * s + 2] - pos[3 * d + 2];
      float ew = sqrtf(rx * rx + ry * ry + rz * rz);
      attw[lane * 8 + 0] = rx;
      attw[lane * 8 + 1] = ry;
      attw[lane * 8 + 2] = rz;
      attw[lane * 8 + 3] = beam[2 * eid];
      attw[lane * 8 + 4] = beam[2 * eid + 1];
      attw[lane * 8 + 5] = ew;
    }
    lds_fence();
    // stage A tile: e (recomputed) + gathered h rows
    for (int idx = lane; idx < 16 * HDIM; idx += 32) {
      int m = idx >> 7, c = idx & 127;
      float v;
      if (c < 64) {
        v = attw[m * 8] * w_e1[c * 3] + attw[m * 8 + 1] * w_e1[c * 3 + 1] +
            attw[m * 8 + 2] * w_e1[c * 3 + 2] + b_e1[c];
      } else {
        int cc = c - 64;
        v = attw[m * 8 + 3] * w_e12[cc * 3] + attw[m * 8 + 4] * w_e12[cc * 3 + 1] +
            attw[m * 8 + 5] * w_e12[cc * 3 + 2] + b_e12[cc];
      }
      sAw[m * APAD + c] = swishf(v);
      int s = srcw[m], d = dstw[m];
      sAw[m * APAD + 128 + c] = h_in[s * HDIM + c];
      sAw[m * APAD + 256 + c] = h_in[d * HDIM + c];
    }
    lds_fence();

    v8f acc[8];
#pragma unroll
    for (int j = 0; j < 8; ++j) {
#pragma unroll
      for (int q = 0; q < 8; ++q) acc[j][q] = 0.0f;
    }
    const float* aRow = sAw + m16 * APAD;
    for (int kb = 0; kb < KIN / 4; ++kb) {
      int kk = kb * 4 + khalf * 2;
      v2f a;
      a.x = aRow[kk];
      a.y = aRow[kk + 1];
#pragma unroll
      for (int nb = 0; nb < 8; ++nb) {
        int n = nb * 16 + m16;
        v2f b;
        b.x = sB[kk * BPAD + n];
        b.y = sB[(kk + 1) * BPAD + n];
        acc[nb] = __builtin_amdgcn_wmma_f32_16x16x4_f32(
            false, a, false, b, (short)0, acc[nb], false, false);
      }
    }
    // epilogue: bias + swish + fused message scatter
#pragma unroll
    for (int nb = 0; nb < 8; ++nb) {
      int n = nb * 16 + m16;
      float bn = bias[n];
#pragma unroll
      for (int r = 0; r < 8; ++r) {
        int m = r + 8 * khalf;
        float wv = swishf(acc[nb][r] + bn);
        float msg = sAw[m * APAD + 128 + n] * wv;
        atomicAdd(&h_out[dstw[m] * HDIM + n], msg);
      }
    }
  }
}

// ---------------------------------------------------------------- node GEMM: y = swish(x @ Bt + bias)
__global__ __launch_bounds__(64) void node_gemm_swish(
    const float* __restrict__ x, float* __restrict__ y,
    const float* __restrict__ Bt, const float* __restrict__ bias) {
  extern __shared__ char smemraw[];
  float* sB = (float*)smemraw;          // HDIM*BPAD
  float* sA = sB + HDIM * BPAD;         // 2*16*APAD2

  const int tid = threadIdx.x;
  const int wave = tid >> 5;
  const int lane = tid & 31;

  for (int i = tid; i < HDIM * HDIM; i += 64) {
    int k = i >> 7, n = i & 127;
    sB[k * BPAD + n] = Bt[i];
  }
  __syncthreads();

  float* sAw = sA + wave * (16 * APAD2);
  const int m16 = lane & 15;
  const int khalf = lane >> 4;
  const int ntiles = N_NODES / 16;

  for (int tile = blockIdx.x * 2 + wave; tile < ntiles; tile += gridDim.x * 2) {
    int base = tile * 16;
    for (int idx = lane; idx < 16 * HDIM; idx += 32) {
      int m = idx >> 7, c = idx & 127;
      sAw[m * APAD2 + c] = x[(base + m) * HDIM + c];
    }
    lds_fence();

    v8f acc[8];
#pragma unroll
    for (int j = 0; j < 8; ++j) {
#pragma unroll
      for (int q = 0; q < 8; ++q) acc[j][q] = 0.0f;
    }
    const float* aRow = sAw + m16 * APAD2;
    for (int kb = 0; kb < HDIM / 4; ++kb) {
      int kk = kb * 4 + khalf * 2;
      v2f a;
      a.x = aRow[kk];
      a.y = aRow[kk + 1];
#pragma unroll
      for (int nb = 0; nb < 8; ++nb) {
        int n = nb * 16 + m16;
        v2f b;
        b.x = sB[kk * BPAD + n];
        b.y = sB[(kk + 1) * BPAD + n];
        acc[nb] = __builtin_amdgcn_wmma_f32_16x16x4_f32(
            false, a, false, b, (short)0, acc[nb], false, false);
      }
    }
#pragma unroll
    for (int nb = 0; nb < 8; ++nb) {
      int n = nb * 16 + m16;
      float bn = bias[n];
#pragma unroll
      for (int r = 0; r < 8; ++r) {
        int m = r + 8 * khalf;
        y[(base + m) * HDIM + n] = swishf(acc[nb][r] + bn);
      }
    }
    lds_fence();
  }
}

// ---------------------------------------------------------------- GraphNorm
__global__ __launch_bounds__(256) void gn_sum_kernel(const float* __restrict__ x,
                                                     float* __restrict__ sums) {
  __shared__ float ls[256];
  const int tid = threadIdx.x;
  ls[tid] = 0.0f;
  __syncthreads();
  const int total = N_NODES * HDIM;
  for (int i = blockIdx.x * blockDim.x + tid; i < total; i += gridDim.x * blockDim.x) {
    float v = x[i];
    int c = i & 127;
    atomicAdd(&ls[c], v);
    atomicAdd(&ls[128 + c], v * v);
  }
  __syncthreads();
  atomicAdd(&sums[tid], ls[tid]);
}

__global__ void gn_apply_kernel(float* __restrict__ x, const float* __restrict__ sums,
                                const float* __restrict__ w, const float* __restrict__ b,
                                const float* __restrict__ ms) {
  const int total = N_NODES * HDIM;
  const float invN = 1.0f / (float)N_NODES;
  for (int i = blockIdx.x * blockDim.x + threadIdx.x; i < total; i += gridDim.x * blockDim.x) {
    int c = i & 127;
    float mean = sums[c] * invN;
    float ex2 = sums[128 + c] * invN;
    float m = ms[c];
    float var = fmaxf(ex2 - m * (2.0f - m) * mean * mean, 0.0f);
    float out = x[i] - m * mean;
    float yn = w[c] * out * __builtin_amdgcn_rsqf(var + EPSN) + b[c];
    x[i] = swishf(yn);
  }
}

// ---------------------------------------------------------------- final energy combine
__global__ void combine_energy_kernel(const float* __restrict__ energy,
                                      const float* __restrict__ w_skip,
                                      const float* __restrict__ b_skip,
                                      float* __restrict__ out) {
  int g = threadIdx.x;
  if (g < N_GRAPH) {
    float s = b_skip[0];
#pragma unroll
    for (int j = 0; j < LLAYERS + 1; ++j) s += energy[g * (LLAYERS + 1) + j] * w_skip[j];
    out[g] = s;
  }
}

// ---------------------------------------------------------------- host launch
extern "C" void kernel_launch(void* const* d_in, const int* in_sizes, int n_in,
                              void* d_out, int out_size, void* d_ws, size_t ws_size,
                              hipStream_t stream) {
  (void)in_sizes; (void)n_in; (void)out_size; (void)ws_size;

  const float* pos    = (const float*)d_in[0];
  const float* forces = (const float*)d_in[1];
  const float* beam   = (const float*)d_in[2];
  const float* w_e1   = (const float*)d_in[3];
  const float* b_e1   = (const float*)d_in[4];
  const float* w_e12  = (const float*)d_in[5];
  const float* b_e12  = (const float*)d_in[6];
  const float* w_h1   = (const float*)d_in[7];
  const float* b_h1   = (const float*)d_in[8];
  const float* w_h12  = (const float*)d_in[9];
  const float* b_h12  = (const float*)d_in[10];
  const float* w_geom = (const float*)d_in[11];
  const float* b_geom = (const float*)d_in[12];
  const float* w_linh = (const float*)d_in[13];
  const float* b_linh = (const float*)d_in[14];
  const float* w_mlp  = (const float*)d_in[15];
  const float* b_mlp  = (const float*)d_in[16];
  const float* gn_w   = (const float*)d_in[17];
  const float* gn_b   = (const float*)d_in[18];
  const float* gn_ms  = (const float*)d_in[19];
  const float* w_lin1 = (const float*)d_in[20];
  const float* b_lin1 = (const float*)d_in[21];
  const float* w_lin2 = (const float*)d_in[22];
  const float* b_lin2 = (const float*)d_in[23];
  const float* w_wlin = (const float*)d_in[24];
  const float* b_wlin = (const float*)d_in[25];
  const float* w_skip = (const float*)d_in[26];
  const float* b_skip = (const float*)d_in[27];
  const int*   eidx   = (const int*)d_in[28];
  const int*   batch  = (const int*)d_in[29];
  const int* esrc = eidx;
  const int* edst = eidx + N_EDGES;

  float* ws = (float*)d_ws;
  const int NH = N_NODES * HDIM;
  float* hA     = ws;
  float* hB     = hA + NH;
  float* sums   = hB + NH;                     // 256
  float* energy = sums + 256;                  // 320
  float* BtG    = energy + N_GRAPH * (LLAYERS + 1);
  float* BtL    = BtG + LLAYERS * KIN * HDIM;
  float* BtM    = BtL + LLAYERS * HDIM * HDIM;
  float* outE   = (float*)d_out;

  const int EDGE_LDS = (KIN * BPAD + 2 * 16 * APAD + 2 * 16 * 8) * 4 + 64 * 4;
  const int NODE_LDS = (HDIM * BPAD + 2 * 16 * APAD2) * 4;

  zero_kernel<<<2, 256, 0, stream>>>(energy, N_GRAPH * (LLAYERS + 1));
  transpose_w_kernel<<<384, 256, 0, stream>>>(BtG, w_geom, LLAYERS, KIN);
  transpose_w_kernel<<<256, 256, 0, stream>>>(BtL, w_linh, LLAYERS, HDIM);
  transpose_w_kernel<<<256, 256, 0, stream>>>(BtM, w_mlp, LLAYERS, HDIM);
  embed_h_kernel<<<2048, 256, 0, stream>>>(forces, w_h1, b_h1, w_h12, b_h12, hA);

  float* cur = hA;
  float* nxt = hB;
  for (int i = 0; i < LLAYERS; ++i) {
    out_block_kernel<<<512, 256, 0, stream>>>(cur, batch, w_wlin, b_wlin, w_lin1, b_lin1,
                                              w_lin2, b_lin2, energy, i);
    zero_kernel<<<2048, 256, 0, stream>>>(nxt, NH);
    edge_gemm_scatter<<<1024, 64, EDGE_LDS, stream>>>(
        cur, nxt, pos, beam, esrc, edst, BtG + i * KIN * HDIM, b_geom + i * HDIM,
        w_e1, b_e1, w_e12, b_e12);
    zero_kernel<<<1, 256, 0, stream>>>(sums, 256);
    gn_sum_kernel<<<1024, 256, 0, stream>>>(nxt, sums);
    gn_apply_kernel<<<2048, 256, 0, stream>>>(nxt, sums, gn_w + i * HDIM, gn_b + i * HDIM,
                                              gn_ms + i * HDIM);
    node_gemm_swish<<<512, 64, NODE_LDS, stream>>>(nxt, cur, BtL + i * HDIM * HDIM,
                                                   b_linh + i * HDIM);
    node_gemm_swish<<<512, 64, NODE_LDS, stream>>>(cur, nxt, BtM + i * HDIM * HDIM,
                                                   b_mlp + i * HDIM);
    float* t = cur; cur = nxt; nxt = t;
  }
  out_block_kernel<<<512, 256, 0, stream>>>(cur, batch, w_wlin, b_wlin, w_lin1, b_lin1,
                                            w_lin2, b_lin2, energy, LLAYERS);
  combine_energy_kernel<<<1, 64, 0, stream>>>(energy, w_skip, b_skip, outE);
  copy_kernel<<<2048, 256, 0, stream>>>(outE + N_GRAPH, cur, NH);
}